// STNet_47596827574464
// MI455X (gfx1250) — compile-verified
//
#include <hip/hip_runtime.h>
#include <hip/hip_bf16.h>
#include <float.h>
#include <math.h>

// STNet reprojection + grid_sample for MI455X (gfx1250, wave32).
// Pass 1: WMMA-projected uv, global max reduction (monotone uint key).
// Pass 2: recompute uv via WMMA, normalize, bilinear gather, store.

#define WG 640                  // grid / image width
#define HG 480                  // grid / image height
#define NPIX (WG * HG)          // 307200 pixels per batch
#define GROUPS (NPIX / 32)      // 9600 groups of 32 pixels

static constexpr int P1_BLOCKS_X = 40;   // 40 blk * 8 waves * 30 iters = 9600
static constexpr int P1_ITERS    = 30;
static constexpr int P2_BLOCKS_X = 80;   // 80 blk * 8 waves * 15 iters = 9600
static constexpr int P2_ITERS    = 15;

typedef float v2f __attribute__((ext_vector_type(2)));
typedef float v8f __attribute__((ext_vector_type(8)));

// ---- order-preserving float <-> uint key for atomic max over signed floats
__device__ __forceinline__ unsigned fkey(float f) {
    unsigned u = __float_as_uint(f);
    return (u & 0x80000000u) ? ~u : (u | 0x80000000u);
}
__device__ __forceinline__ float fdec(unsigned k) {
    unsigned u = (k & 0x80000000u) ? (k & 0x7FFFFFFFu) : ~k;
    return __uint_as_float(u);
}

// ---- fast reciprocal: v_rcp_f32 + one Newton-Raphson step (~1 ulp)
__device__ __forceinline__ float fast_rcp(float t) {
    float r = __builtin_amdgcn_rcpf(t);
    return r * (2.0f - t * r);
}

// ---- H = [[K^-1 * T[:3,:3] * K], [T[3,:3] * K]]  (4x3), row-vector convention
__device__ void computeH(const float* __restrict__ T, const float* __restrict__ Km,
                         float Hh[4][3]) {
    float k[9];
#pragma unroll
    for (int i = 0; i < 9; ++i) k[i] = Km[i];
    // general 3x3 inverse (adjugate / det)
    float c00 =  k[4] * k[8] - k[5] * k[7];
    float c01 = -(k[3] * k[8] - k[5] * k[6]);
    float c02 =  k[3] * k[7] - k[4] * k[6];
    float det = k[0] * c00 + k[1] * c01 + k[2] * c02;
    float id  = 1.0f / det;   // once per kernel, precise path is fine
    float ki[9];
    ki[0] = c00 * id;
    ki[1] = -(k[1] * k[8] - k[2] * k[7]) * id;
    ki[2] =  (k[1] * k[5] - k[2] * k[4]) * id;
    ki[3] = c01 * id;
    ki[4] =  (k[0] * k[8] - k[2] * k[6]) * id;
    ki[5] = -(k[0] * k[5] - k[2] * k[3]) * id;
    ki[6] = c02 * id;
    ki[7] = -(k[0] * k[7] - k[1] * k[6]) * id;
    ki[8] =  (k[0] * k[4] - k[1] * k[3]) * id;
    // M[i][j] = sum_l T[i][l] * K[l][j], i = 0..3 (row 3 -> translation t)
    float M[4][3];
#pragma unroll
    for (int i = 0; i < 4; ++i)
#pragma unroll
        for (int j = 0; j < 3; ++j)
            M[i][j] = T[i * 4 + 0] * k[0 * 3 + j] +
                      T[i * 4 + 1] * k[1 * 3 + j] +
                      T[i * 4 + 2] * k[2 * 3 + j];
    // N = K^-1 @ M[0:3]
#pragma unroll
    for (int i = 0; i < 3; ++i)
#pragma unroll
        for (int j = 0; j < 3; ++j)
            Hh[i][j] = ki[i * 3 + 0] * M[0][j] +
                       ki[i * 3 + 1] * M[1][j] +
                       ki[i * 3 + 2] * M[2][j];
#pragma unroll
    for (int j = 0; j < 3; ++j) Hh[3][j] = M[3][j];
}

// ---- A operand: 16x4 f32 matrix, rows {0,1,2} and {8,9,10} = H^T, rest zero.
// ISA layout: lanes 0-15 hold row M=lane (K=0 in v0, K=1 in v1);
//             lanes 16-31 hold row M=lane-16 (K=2 in v0, K=3 in v1).
__device__ __forceinline__ v2f buildA(const float Hh[4][3], int lane) {
    int mm  = lane & 15;
    int col = (mm < 3) ? mm : ((mm >= 8 && mm < 11) ? (mm - 8) : -1);
    v2f a;
    a.x = 0.0f;
    a.y = 0.0f;
    if (col >= 0) {
        if (lane < 16) { a.x = Hh[0][col]; a.y = Hh[1][col]; }
        else           { a.x = Hh[2][col]; a.y = Hh[3][col]; }
    }
    return a;
}

// ---- project 32 pixels (pbase..pbase+31) -> uv for this lane's pixel pbase+lane
__device__ __forceinline__ void project32(const v2f aOp, const float Hh[4][3],
                                          const float* __restrict__ depthB,
                                          int pbase, int lane, bool lo,
                                          float& ux, float& uy) {
    int i1 = pbase + (lane & 15);
    int i2 = i1 + 16;
    float d1v = depthB[i1];
    float d2v = depthB[i2];
    int w1 = i1 / HG, h1i = i1 - w1 * HG;
    int w2 = i2 / HG, h2i = i2 - w2 * HG;
    // B operand: 4x16 f32, column n = pixel vector (h*d, w*d, d, 1).
    // Assumed layout mirrors A: lanes 0-15 supply K=0(v0),K=1(v1); lanes 16-31 K=2,K=3.
    // Branchless: compute both halves' values, select per component (v_cndmask).
    float b1x_lo = (float)h1i * d1v, b1y_lo = (float)w1 * d1v;
    float b2x_lo = (float)h2i * d2v, b2y_lo = (float)w2 * d2v;
    v2f b1, b2;
    b1.x = lo ? b1x_lo : d1v;
    b1.y = lo ? b1y_lo : 1.0f;
    b2.x = lo ? b2x_lo : d2v;
    b2.y = lo ? b2y_lo : 1.0f;
    float qx, qy, qz;
#if __has_builtin(__builtin_amdgcn_wmma_f32_16x16x4_f32)
    v8f z = {};
    // D = A(16x4) x B(4x16): rows 0..2 -> pixels of B1 at lanes 0-15 (v0..v2),
    // rows 8..10 -> pixels of B2 at lanes 16-31 (v0..v2).
    v8f dA = __builtin_amdgcn_wmma_f32_16x16x4_f32(
        false, aOp, false, b1, (short)0, z, false, false);
    v8f dB = __builtin_amdgcn_wmma_f32_16x16x4_f32(
        false, aOp, false, b2, (short)0, z, false, false);
    qx = lo ? dA[0] : dB[0];
    qy = lo ? dA[1] : dB[1];
    qz = lo ? dA[2] : dB[2];
#else
    // VALU fallback: q = (h*d, w*d, d, 1) @ H for own pixel
    int ip = pbase + lane;
    int wp = ip / HG, hp = ip - wp * HG;
    float dp = depthB[ip];
    float p0 = (float)hp * dp, p1 = (float)wp * dp;
    qx = p0 * Hh[0][0] + p1 * Hh[1][0] + dp * Hh[2][0] + Hh[3][0];
    qy = p0 * Hh[0][1] + p1 * Hh[1][1] + dp * Hh[2][1] + Hh[3][1];
    qz = p0 * Hh[0][2] + p1 * Hh[1][2] + dp * Hh[2][2] + Hh[3][2];
#endif
    float inv = fast_rcp(qz + 1e-4f);
    ux = qx * inv;
    uy = qy * inv;
}

// ================= Pass 1: global max of uv =================
__global__ __launch_bounds__(256) void stnet_pass1(
    const float* __restrict__ depth, const float* __restrict__ T,
    const float* __restrict__ K, unsigned* __restrict__ gmax) {
    __shared__ float smax[8];
    int b = blockIdx.y;
    float Hh[4][3];
    computeH(T + b * 16, K, Hh);
    int lane = threadIdx.x & 31;
    bool lo = lane < 16;
    v2f aOp = buildA(Hh, lane);
    const float* depthB = depth + (size_t)b * NPIX;
    int waveId = blockIdx.x * (blockDim.x >> 5) + (threadIdx.x >> 5);
    float mx = -FLT_MAX;
    for (int it = 0; it < P1_ITERS; ++it) {
        int group = waveId * P1_ITERS + it;
        float ux, uy;
        project32(aOp, Hh, depthB, group * 32, lane, lo, ux, uy);
        mx = fmaxf(mx, fmaxf(ux, uy));
    }
#pragma unroll
    for (int off = 16; off > 0; off >>= 1)
        mx = fmaxf(mx, __shfl_xor(mx, off, 32));
    if (lane == 0) smax[threadIdx.x >> 5] = mx;
    __syncthreads();
    if (threadIdx.x == 0) {
        float m = smax[0];
#pragma unroll
        for (int i = 1; i < 8; ++i) m = fmaxf(m, smax[i]);
        atomicMax(gmax, fkey(m));
    }
}

// ================= Pass 2: normalize + bilinear grid_sample =================
__global__ __launch_bounds__(256) void stnet_pass2(
    const float* __restrict__ depth, const float* __restrict__ T,
    const float* __restrict__ image, const float* __restrict__ K,
    const unsigned* __restrict__ gmax, float* __restrict__ out) {
    int b = blockIdx.y;
    float Hh[4][3];
    computeH(T + b * 16, K, Hh);
    int lane = threadIdx.x & 31;
    bool lo = lane < 16;
    v2f aOp = buildA(Hh, lane);
    const float* depthB = depth + (size_t)b * NPIX;
    const float* imgB = image + (size_t)b * 3 * NPIX;
    float* outB = out + (size_t)b * 3 * NPIX;
    float invmax = 1.0f / fdec(*gmax);   // once per thread, precise
    int waveId = blockIdx.x * (blockDim.x >> 5) + (threadIdx.x >> 5);
    for (int it = 0; it < P2_ITERS; ++it) {
        int group = waveId * P2_ITERS + it;
        int pbase = group * 32;
        float ux, uy;
        project32(aOp, Hh, depthB, pbase, lane, lo, ux, uy);
        float uxn = 2.0f * ux * invmax - 1.0f;
        float uyn = 2.0f * uy * invmax - 1.0f;
        // grid_sample, align_corners=False, padding zeros; x uses W=640, y uses H=480
        float x = (uxn + 1.0f) * (WG * 0.5f) - 0.5f;
        float y = (uyn + 1.0f) * (HG * 0.5f) - 0.5f;
        float x0 = floorf(x), y0 = floorf(y);
        float wx = x - x0, wy = y - y0;
        float a0 = 0.0f, a1 = 0.0f, a2 = 0.0f;
#pragma unroll
        for (int c4 = 0; c4 < 4; ++c4) {
            float xc = x0 + (float)(c4 & 1);
            float yc = y0 + (float)(c4 >> 1);
            float wgt = ((c4 & 1) ? wx : 1.0f - wx) * ((c4 >> 1) ? wy : 1.0f - wy);
            bool valid = (xc >= 0.0f) & (xc <= (float)(WG - 1)) &
                         (yc >= 0.0f) & (yc <= (float)(HG - 1));
            float m = valid ? wgt : 0.0f;
            int xi = (int)fminf(fmaxf(xc, 0.0f), (float)(WG - 1));
            int yi = (int)fminf(fmaxf(yc, 0.0f), (float)(HG - 1));
            int off = yi * WG + xi;
            a0 += m * imgB[off];
            a1 += m * imgB[NPIX + off];
            a2 += m * imgB[2 * NPIX + off];
        }
        int i = pbase + lane;                 // output [B,3,640,480]: i = w*480+h
        outB[i] = a0;
        outB[NPIX + i] = a1;
        outB[2 * NPIX + i] = a2;
    }
}

extern "C" void kernel_launch(void* const* d_in, const int* in_sizes, int n_in,
                              void* d_out, int out_size, void* d_ws, size_t ws_size,
                              hipStream_t stream) {
    const float* depth = (const float*)d_in[0];   // [B,640,480,1]
    const float* T     = (const float*)d_in[1];   // [B,4,4]
    const float* image = (const float*)d_in[2];   // [B,3,480,640]
    const float* K     = (const float*)d_in[3];   // [3,3]
    int B = in_sizes[0] / NPIX;

    hipMemsetAsync(d_ws, 0, sizeof(unsigned), stream);  // key 0 < any real key

    dim3 blk(256);
    dim3 g1(P1_BLOCKS_X, B);
    stnet_pass1<<<g1, blk, 0, stream>>>(depth, T, K, (unsigned*)d_ws);
    dim3 g2(P2_BLOCKS_X, B);
    stnet_pass2<<<g2, blk, 0, stream>>>(depth, T, image, K,
                                        (const unsigned*)d_ws, (float*)d_out);
}